// ContxE_30631706755343
// MI455X (gfx1250) — compile-verified
//
#include <hip/hip_runtime.h>
#include <stdint.h>

// ContxE scoring kernel for MI455X (gfx1250).
// Memory-bound gather workload: ~369 MB of row gathers -> ~16 us HBM floor,
// mostly L2-resident in steady state. Uses CDNA5 async global->LDS DMA
// (global_load_async_to_lds_b128 / ASYNCcnt) for the gathers, one wave32 per
// batch item, wave-shuffle butterfly reductions. WMMA intentionally unused:
// the contraction is a batched per-item dot with no cross-item operand
// sharing (6% tile utilization if forced through 16x16 WMMA).

#define DIM        512
#define BATCH      16384
#define N_DAY      365
#define WIN        5

#define WAVES_PER_BLOCK 2
#define ROWS_PER_WAVE   11   // 0:hr 1:hi 2:tr 3:ti 4:rr 5:ri 6..10:time[w]

struct V16 { float v[16]; };

// One 2KB row = 4 async b128 ops per wave32 (32 lanes x 16B = 512B each).
// The instruction's offset: immediate is added to BOTH the global address and
// the LDS address (CDNA5 ISA), so one address setup covers
// the whole row.
__device__ __forceinline__ void async_row_load(uint32_t lds_byte, uint64_t gaddr) {
  asm volatile(
      "global_load_async_to_lds_b128 %0, %1, off\n\t"
      "global_load_async_to_lds_b128 %0, %1, off offset:512\n\t"
      "global_load_async_to_lds_b128 %0, %1, off offset:1024\n\t"
      "global_load_async_to_lds_b128 %0, %1, off offset:1536"
      :: "v"(lds_byte), "v"(gaddr)
      : "memory");
}

__device__ __forceinline__ V16 loadRow16(const float* p) {
  V16 r;
  const float4* q = (const float4*)p;
#pragma unroll
  for (int i = 0; i < 4; ++i) {
    float4 t = q[i];
    r.v[4*i+0] = t.x; r.v[4*i+1] = t.y; r.v[4*i+2] = t.z; r.v[4*i+3] = t.w;
  }
  return r;
}

__device__ __forceinline__ float wave_sum(float x) {
#pragma unroll
  for (int m = 16; m >= 1; m >>= 1) x += __shfl_xor(x, m, 32);
  return x;
}

__device__ __forceinline__ void softmax5(float* x) {
  float m = x[0];
#pragma unroll
  for (int w = 1; w < WIN; ++w) m = fmaxf(m, x[w]);
  float s = 0.f;
#pragma unroll
  for (int w = 0; w < WIN; ++w) { x[w] = __expf(x[w] - m); s += x[w]; }
  float inv = 1.0f / s;
#pragma unroll
  for (int w = 0; w < WIN; ++w) x[w] *= inv;
}

__global__ __launch_bounds__(WAVES_PER_BLOCK * 32)
void contxe_kernel(const int* __restrict__ h_i, const int* __restrict__ t_i,
                   const int* __restrict__ r_i, const int* __restrict__ d_i,
                   const float* __restrict__ Er, const float* __restrict__ Ei,
                   const float* __restrict__ Rr, const float* __restrict__ Ri,
                   const float* __restrict__ T,  float* __restrict__ out) {
  __shared__ float smem[WAVES_PER_BLOCK * ROWS_PER_WAVE * DIM];  // 44 KB

  const int lane = threadIdx.x & 31;
  const int wave = threadIdx.x >> 5;
  const int item = blockIdx.x * WAVES_PER_BLOCK + wave;

  // Wave-uniform indices -> force SGPR.
  const int h  = __builtin_amdgcn_readfirstlane(h_i[item]);
  const int t  = __builtin_amdgcn_readfirstlane(t_i[item]);
  const int r  = __builtin_amdgcn_readfirstlane(r_i[item]);
  const int d0 = __builtin_amdgcn_readfirstlane(d_i[item]);

  float* myLds = smem + wave * (ROWS_PER_WAVE * DIM);
  // Low 32 bits of the generic shared pointer == wave-relative LDS byte
  // offset (ISA 10.2: LDS aperture truncates addr[31:0]).
  const uint32_t ldsBase  = (uint32_t)(uintptr_t)(void*)myLds;
  const uint32_t laneByte = (uint32_t)lane * 16u;

  // --- async gather: 11 rows (22.5 KB), 44 requests in flight ---
  {
    const float* rows[ROWS_PER_WAVE];
    rows[0] = Er + (size_t)h * DIM;
    rows[1] = Ei + (size_t)h * DIM;
    rows[2] = Er + (size_t)t * DIM;
    rows[3] = Ei + (size_t)t * DIM;
    rows[4] = Rr + (size_t)r * DIM;
    rows[5] = Ri + (size_t)r * DIM;
#pragma unroll
    for (int w = 0; w < WIN; ++w) {
      int idx = d0 - (WIN - 1 - w);
      idx = (idx >= 0) ? idx : N_DAY;       // negatives -> pad slot n_day
      rows[6 + w] = T + (size_t)idx * DIM;
    }
#pragma unroll
    for (int s = 0; s < ROWS_PER_WAVE; ++s) {
      async_row_load(ldsBase + (uint32_t)s * (DIM * 4) + laneByte,
                     (uint64_t)(uintptr_t)rows[s] + laneByte);
    }
  }
  asm volatile("s_wait_asynccnt 0x0" ::: "memory");

  const float* L = myLds + lane * 16;
  const V16 hr = loadRow16(L + 0 * DIM);
  const V16 hi = loadRow16(L + 1 * DIM);
  const V16 tr = loadRow16(L + 2 * DIM);
  const V16 ti = loadRow16(L + 3 * DIM);
  const V16 rr = loadRow16(L + 4 * DIM);
  const V16 ri = loadRow16(L + 5 * DIM);

  // --- pass 1: windowed attention scores (4 dots per w, wave-reduced) ---
  float sA[WIN], sB[WIN], sC[WIN], sD[WIN];
#pragma unroll
  for (int w = 0; w < WIN; ++w) {
    V16 tv = loadRow16(L + (6 + w) * DIM);
    float a = 0.f, b = 0.f, c = 0.f, d = 0.f;
#pragma unroll
    for (int j = 0; j < 16; ++j) {
      float sn = __sinf(tv.v[j]);
      float cs = __cosf(tv.v[j]);
      float hre = fmaf(hr.v[j], cs, -hi.v[j] * sn);
      float him = fmaf(hr.v[j], sn,  hi.v[j] * cs);
      float tre = fmaf(tr.v[j], cs, -ti.v[j] * sn);
      float tim = fmaf(tr.v[j], sn,  ti.v[j] * cs);
      a = fmaf(rr.v[j], hre, a);
      b = fmaf(ri.v[j], him, b);
      c = fmaf(rr.v[j], tre, c);
      d = fmaf(ri.v[j], tim, d);
    }
    sA[w] = wave_sum(a);
    sB[w] = wave_sum(b);
    sC[w] = wave_sum(c);
    sD[w] = wave_sum(d);
  }
  softmax5(sA);  // a_real
  softmax5(sB);  // a_img
  softmax5(sC);  // b_real
  softmax5(sD);  // b_img

  // --- pass 2: recompute rotations (cheap TRANS ops), fold y-z directly ---
  float dr[16], di[16];
#pragma unroll
  for (int j = 0; j < 16; ++j) { dr[j] = 0.f; di[j] = 0.f; }
#pragma unroll
  for (int w = 0; w < WIN; ++w) {
    V16 tv = loadRow16(L + (6 + w) * DIM);
    const float aR = sA[w], aI = sB[w], bR = sC[w], bI = sD[w];
#pragma unroll
    for (int j = 0; j < 16; ++j) {
      float sn = __sinf(tv.v[j]);
      float cs = __cosf(tv.v[j]);
      float hre = fmaf(hr.v[j], cs, -hi.v[j] * sn);
      float him = fmaf(hr.v[j], sn,  hi.v[j] * cs);
      float tre = fmaf(tr.v[j], cs, -ti.v[j] * sn);
      float tim = fmaf(tr.v[j], sn,  ti.v[j] * cs);
      dr[j] = fmaf(aR, hre, fmaf(-bR, tre, dr[j]));  // y_real - z_real
      di[j] = fmaf(aI, him, fmaf( bI, tim, di[j]));  // y_img  + z_img
    }
  }

  float acc = 0.f;
#pragma unroll
  for (int j = 0; j < 16; ++j) {
    acc += fabsf(dr[j] + rr.v[j]);   // |y_real + r_real - z_real|
    acc += fabsf(di[j] + ri.v[j]);   // |y_img  + r_img  + z_img|
  }
  acc = wave_sum(acc);
  if (lane == 0) out[item] = acc;
}

extern "C" void kernel_launch(void* const* d_in, const int* in_sizes, int n_in,
                              void* d_out, int out_size, void* d_ws, size_t ws_size,
                              hipStream_t stream) {
  (void)in_sizes; (void)n_in; (void)d_ws; (void)ws_size; (void)out_size;
  const int*   h_i = (const int*)d_in[0];
  const int*   t_i = (const int*)d_in[1];
  const int*   r_i = (const int*)d_in[2];
  const int*   d_i = (const int*)d_in[3];
  const float* Er  = (const float*)d_in[4];
  const float* Ei  = (const float*)d_in[5];
  const float* Rr  = (const float*)d_in[6];
  const float* Ri  = (const float*)d_in[7];
  const float* T   = (const float*)d_in[8];
  float* out = (float*)d_out;

  dim3 grid(BATCH / WAVES_PER_BLOCK);
  dim3 block(WAVES_PER_BLOCK * 32);
  contxe_kernel<<<grid, block, 0, stream>>>(h_i, t_i, r_i, d_i, Er, Ei, Rr, Ri, T, out);
}